// InferenceBox_20529943675129
// MI455X (gfx1250) — compile-verified
//
#include <hip/hip_runtime.h>

// ---------------- problem constants (from reference) ----------------
#define BATCH    64
#define NBOX     24564
#define NCH      85          // 4 loc + 81 conf
#define NCONF    81
#define LOC_VAR  0.1f
#define SIZE_VAR 0.2f
#define THRESH   0.01f

// ---------------- tiling ----------------
#define ROWS_PER_BLOCK 96    // 96*85 floats = 32640 B LDS, 16B-divisible
#define THREADS        256   // 8 wave32 waves

typedef float v4f __attribute__((ext_vector_type(4)));
typedef int   v4i __attribute__((ext_vector_type(4)));

// Builtin signature (from hipcc diagnostic): first param is
// 'int __vector(4) __device__ *'  ==  addrspace(1) v4i*.
typedef __attribute__((address_space(1))) v4i g_v4i_t;
typedef __attribute__((address_space(3))) v4i l_v4i_t;

// Address-space casts via integer round-trip:
//  - flat LDS pointer keeps the wave-relative LDS byte offset in addr[31:0]
//    (ISA 10.2),
//  - flat global pointer is numerically the global VA.
__device__ __forceinline__ g_v4i_t* to_glb4(const void* p) {
    return (g_v4i_t*)(unsigned long long)p;
}
__device__ __forceinline__ l_v4i_t* to_lds4(const void* p) {
    return (l_v4i_t*)(unsigned)(unsigned long long)p;
}

#if __has_builtin(__builtin_amdgcn_global_load_async_to_lds_b128) && \
    __has_builtin(__builtin_amdgcn_s_wait_asynccnt)
#define HAVE_ASYNC_LDS 1
#else
#define HAVE_ASYNC_LDS 0
#endif

__global__ __launch_bounds__(THREADS)
void InferenceBox_decode_kernel(const float* __restrict__ predicts,
                                const float* __restrict__ dboxes,
                                float* __restrict__ out)
{
    __shared__ float smem[ROWS_PER_BLOCK * NCH];   // 32640 bytes

    const int b    = blockIdx.y;
    const int row0 = blockIdx.x * ROWS_PER_BLOCK;
    const int rows = (NBOX - row0) < ROWS_PER_BLOCK ? (NBOX - row0) : ROWS_PER_BLOCK;
    const int tid  = threadIdx.x;

    // ---- stage one contiguous chunk of `rows` full rows into LDS ----
    // base byte offset = (b*NBOX + row0)*340 : multiple of 16, so b128 ok.
    const size_t in_base = ((size_t)b * NBOX + (size_t)row0) * NCH;
    const v4f* __restrict__ src4 = (const v4f*)(predicts + in_base);
    const int nvec4 = rows * NCH / 4;              // rows%4==0 -> exact

#if HAVE_ASYNC_LDS
    for (int v = tid; v < nvec4; v += THREADS) {
        // 16 B per lane, global -> LDS, tracked by ASYNCcnt
        __builtin_amdgcn_global_load_async_to_lds_b128(
            to_glb4(src4 + v), to_lds4(&smem[v * 4]), 0, 0);
    }
    __builtin_amdgcn_s_wait_asynccnt(0);
#else
    {
        v4f* s4 = (v4f*)smem;
        for (int v = tid; v < nvec4; v += THREADS)
            s4[v] = __builtin_nontemporal_load(src4 + v);
    }
#endif
    __syncthreads();

    // ---- output layout: [loc : 64*NBOX*4][indicator : 64*NBOX*81] ----
    float* __restrict__ out_loc  = out;
    float* __restrict__ out_conf = out + (size_t)BATCH * NBOX * 4;

    // ---- box decode: one thread per row (rows <= 96 < THREADS) ----
    if (tid < rows) {
        const int r = tid;
        const float p0 = smem[r * NCH + 0];
        const float p1 = smem[r * NCH + 1];
        const float p2 = smem[r * NCH + 2];
        const float p3 = smem[r * NCH + 3];

        const v4f d = ((const v4f*)dboxes)[row0 + r];   // (cx, cy, w, h), 16B aligned

        const float cx = d.x + p0 * LOC_VAR * d.z;
        const float cy = d.y + p1 * LOC_VAR * d.w;
        const float ww = d.z * expf(p2 * SIZE_VAR);
        const float hh = d.w * expf(p3 * SIZE_VAR);

        v4f o;
        o.x = cx - 0.5f * ww;
        o.y = cy - 0.5f * hh;
        o.z = cx + 0.5f * ww;
        o.w = cy + 0.5f * hh;
        __builtin_nontemporal_store(o, (v4f*)out_loc + ((size_t)b * NBOX + row0 + r));
    }

    // ---- indicator: contiguous rows*81 floats, b128 NT stores ----
    const int nconf4 = rows * NCONF / 4;           // rows%4==0 -> exact
    v4f* __restrict__ outc4 =
        (v4f*)(out_conf + ((size_t)b * NBOX + (size_t)row0) * NCONF);

    for (int v = tid; v < nconf4; v += THREADS) {
        const int e = 4 * v;
        v4f o;
#pragma unroll
        for (int k = 0; k < 4; ++k) {
            const int ee = e + k;
            const int r  = ee / NCONF;             // const-div -> mul/shift
            const int j  = ee - r * NCONF;
            o[k] = smem[r * NCH + 4 + j] > THRESH ? 1.0f : 0.0f;
        }
        __builtin_nontemporal_store(o, outc4 + v);
    }
}

extern "C" void kernel_launch(void* const* d_in, const int* in_sizes, int n_in,
                              void* d_out, int out_size, void* d_ws, size_t ws_size,
                              hipStream_t stream)
{
    (void)in_sizes; (void)n_in; (void)out_size; (void)d_ws; (void)ws_size;

    const float* predicts = (const float*)d_in[0];   // (64, 24564, 85) f32
    const float* dboxes   = (const float*)d_in[1];   // (24564, 4)      f32
    float* out = (float*)d_out;                      // loc(64*24564*4) ++ ind(64*24564*81)

    dim3 grid((NBOX + ROWS_PER_BLOCK - 1) / ROWS_PER_BLOCK, BATCH, 1);  // 256 x 64
    InferenceBox_decode_kernel<<<grid, dim3(THREADS, 1, 1), 0, stream>>>(
        predicts, dboxes, out);
}